// LatentTexture_71777493450866
// MI455X (gfx1250) — compile-verified
//
#include <hip/hip_runtime.h>
#include <stdint.h>

// ---------------- problem constants (match reference shapes) ----------------
#define C_HI   12
#define C_LO   4
#define W_HI   2048
#define H_HI   2048
#define W_LO   512
#define H_LO   512
#define HW_HI  (W_HI * H_HI)
#define HW_LO  (W_LO * H_LO)
#define QMAX_HI 255.0f
#define QMAX_LO 15.0f

#define BLOCK 256

// CDNA5 async global->LDS path (gfx1250). Guarded so compile never breaks.
#if __has_builtin(__builtin_amdgcn_global_load_async_to_lds_b64) && \
    __has_builtin(__builtin_amdgcn_s_wait_asynccnt)
#define USE_ASYNC_UV 1
typedef int v2i __attribute__((vector_size(8)));
typedef __attribute__((address_space(1))) v2i* g_v2i_ptr;
typedef __attribute__((address_space(3))) v2i* l_v2i_ptr;
#else
#define USE_ASYNC_UV 0
#endif

__device__ __forceinline__ int imin(int a, int b) { return a < b ? a : b; }
__device__ __forceinline__ int imax(int a, int b) { return a > b ? a : b; }

// forward of quantize_ste: round(clip(x,0,1)*qmax)/qmax ; rintf == round-half-even
__device__ __forceinline__ float qforward(float x, float qmax, float invq) {
  return rintf(fminf(fmaxf(x, 0.0f), 1.0f) * qmax) * invq;
}

__device__ __forceinline__ unsigned pick_dw(const uint4& t, int i) {
  return i == 0 ? t.x : (i == 1 ? t.y : (i == 2 ? t.z : t.w));
}
__device__ __forceinline__ float ubyte_f(unsigned w, int b) {
  return (float)((w >> (8 * b)) & 0xFFu);   // -> v_cvt_f32_ubyteN
}

// ---------------- pass 1a: quantize+repack hi texture to u8, channel-last, 16B/texel
__global__ __launch_bounds__(BLOCK) void repack_hi_kernel(
    const float* __restrict__ src, uint4* __restrict__ dst) {
  int t = blockIdx.x * BLOCK + threadIdx.x;
  if (t >= HW_HI) return;
  unsigned d0 = 0, d1 = 0, d2 = 0;
#pragma unroll
  for (int c = 0; c < C_HI; ++c) {
    float x = src[(size_t)c * HW_HI + t];
    unsigned k = (unsigned)rintf(fminf(fmaxf(x, 0.0f), 1.0f) * QMAX_HI);
    if (c < 4)       d0 |= k << (8 * (c & 3));
    else if (c < 8)  d1 |= k << (8 * (c & 3));
    else             d2 |= k << (8 * (c & 3));
  }
  dst[t] = make_uint4(d0, d1, d2, 0u);
}

// ---------------- pass 1b: quantize+repack lo texture to u8, channel-last, 4B/texel
__global__ __launch_bounds__(BLOCK) void repack_lo_kernel(
    const float* __restrict__ src, unsigned* __restrict__ dst) {
  int t = blockIdx.x * BLOCK + threadIdx.x;
  if (t >= HW_LO) return;
  unsigned d = 0;
#pragma unroll
  for (int c = 0; c < C_LO; ++c) {
    float x = src[(size_t)c * HW_LO + t];
    unsigned k = (unsigned)rintf(fminf(fmaxf(x, 0.0f), 1.0f) * QMAX_LO);
    d |= k << (8 * c);
  }
  dst[t] = d;
}

// ---------------- pass 2: bilinear gather from packed u8 textures ----------------
__global__ __launch_bounds__(BLOCK) void sample_packed_kernel(
    const float2* __restrict__ uv, const uint4* __restrict__ hi8,
    const unsigned* __restrict__ lo8, float* __restrict__ out, int B) {
  const int tid = threadIdx.x;
  const int gid = blockIdx.x * BLOCK + tid;

  __shared__ float2 suv[BLOCK];
#if USE_ASYNC_UV
  if (gid < B) {
    __builtin_amdgcn_global_load_async_to_lds_b64(
        (g_v2i_ptr)(uv + gid),
        (l_v2i_ptr)(&suv[tid]),
        0, 0);
  }
  __builtin_amdgcn_s_wait_asynccnt(0);
#else
  if (gid < B) suv[tid] = uv[gid];
#endif
  if (gid >= B) return;

  const float u = suv[tid].x;
  const float v = suv[tid].y;

  float r[16];

  // ---- hi texture: align_corners bilinear; 4 aligned B128 gathers ----
  {
    float ix = u * (float)(W_HI - 1);
    float iy = v * (float)(H_HI - 1);
    int x0 = imin(imax((int)floorf(ix), 0), W_HI - 2);
    int y0 = imin(imax((int)floorf(iy), 0), H_HI - 2);
    float wx = fminf(fmaxf(ix - (float)x0, 0.0f), 1.0f);
    float wy = fminf(fmaxf(iy - (float)y0, 0.0f), 1.0f);
    float w00 = (1.0f - wy) * (1.0f - wx);
    float w01 = (1.0f - wy) * wx;
    float w10 = wy * (1.0f - wx);
    float w11 = wy * wx;
    int i00 = y0 * W_HI + x0;
    uint4 t00 = hi8[i00];
    uint4 t01 = hi8[i00 + 1];
    uint4 t10 = hi8[i00 + W_HI];
    uint4 t11 = hi8[i00 + W_HI + 1];
    const float s = 1.0f / QMAX_HI;
#pragma unroll
    for (int c = 0; c < C_HI; ++c) {
      int dw = c >> 2, b = c & 3;
      float a00 = ubyte_f(pick_dw(t00, dw), b);
      float a01 = ubyte_f(pick_dw(t01, dw), b);
      float a10 = ubyte_f(pick_dw(t10, dw), b);
      float a11 = ubyte_f(pick_dw(t11, dw), b);
      r[c] = (w00 * a00 + w01 * a01 + w10 * a10 + w11 * a11) * s;
    }
  }

  // ---- lo texture: 4 B32 gathers ----
  {
    float ix = u * (float)(W_LO - 1);
    float iy = v * (float)(H_LO - 1);
    int x0 = imin(imax((int)floorf(ix), 0), W_LO - 2);
    int y0 = imin(imax((int)floorf(iy), 0), H_LO - 2);
    float wx = fminf(fmaxf(ix - (float)x0, 0.0f), 1.0f);
    float wy = fminf(fmaxf(iy - (float)y0, 0.0f), 1.0f);
    float w00 = (1.0f - wy) * (1.0f - wx);
    float w01 = (1.0f - wy) * wx;
    float w10 = wy * (1.0f - wx);
    float w11 = wy * wx;
    int j00 = y0 * W_LO + x0;
    unsigned s00 = lo8[j00];
    unsigned s01 = lo8[j00 + 1];
    unsigned s10 = lo8[j00 + W_LO];
    unsigned s11 = lo8[j00 + W_LO + 1];
    const float s = 1.0f / QMAX_LO;
#pragma unroll
    for (int c = 0; c < C_LO; ++c) {
      float a00 = ubyte_f(s00, c);
      float a01 = ubyte_f(s01, c);
      float a10 = ubyte_f(s10, c);
      float a11 = ubyte_f(s11, c);
      r[C_HI + c] = (w00 * a00 + w01 * a01 + w10 * a10 + w11 * a11) * s;
    }
  }

  float4* o = (float4*)(out + (size_t)gid * 16);
  o[0] = make_float4(r[0], r[1], r[2], r[3]);
  o[1] = make_float4(r[4], r[5], r[6], r[7]);
  o[2] = make_float4(r[8], r[9], r[10], r[11]);
  o[3] = make_float4(r[12], r[13], r[14], r[15]);
}

// ---------------- fallback: direct gather from f32 planes (if ws too small) ----------------
__global__ __launch_bounds__(BLOCK) void sample_naive_kernel(
    const float2* __restrict__ uv, const float* __restrict__ hi,
    const float* __restrict__ lo, float* __restrict__ out, int B) {
  int gid = blockIdx.x * BLOCK + threadIdx.x;
  if (gid >= B) return;
  float u = uv[gid].x, v = uv[gid].y;
  float r[16];
  {
    float ix = u * (float)(W_HI - 1), iy = v * (float)(H_HI - 1);
    int x0 = imin(imax((int)floorf(ix), 0), W_HI - 2);
    int y0 = imin(imax((int)floorf(iy), 0), H_HI - 2);
    float wx = fminf(fmaxf(ix - (float)x0, 0.0f), 1.0f);
    float wy = fminf(fmaxf(iy - (float)y0, 0.0f), 1.0f);
    float w00 = (1 - wy) * (1 - wx), w01 = (1 - wy) * wx, w10 = wy * (1 - wx), w11 = wy * wx;
    size_t i00 = (size_t)y0 * W_HI + x0;
    const float invq = 1.0f / QMAX_HI;
#pragma unroll
    for (int c = 0; c < C_HI; ++c) {
      const float* p = hi + (size_t)c * HW_HI + i00;
      float a00 = qforward(p[0], QMAX_HI, invq);
      float a01 = qforward(p[1], QMAX_HI, invq);
      float a10 = qforward(p[W_HI], QMAX_HI, invq);
      float a11 = qforward(p[W_HI + 1], QMAX_HI, invq);
      r[c] = w00 * a00 + w01 * a01 + w10 * a10 + w11 * a11;
    }
  }
  {
    float ix = u * (float)(W_LO - 1), iy = v * (float)(H_LO - 1);
    int x0 = imin(imax((int)floorf(ix), 0), W_LO - 2);
    int y0 = imin(imax((int)floorf(iy), 0), H_LO - 2);
    float wx = fminf(fmaxf(ix - (float)x0, 0.0f), 1.0f);
    float wy = fminf(fmaxf(iy - (float)y0, 0.0f), 1.0f);
    float w00 = (1 - wy) * (1 - wx), w01 = (1 - wy) * wx, w10 = wy * (1 - wx), w11 = wy * wx;
    size_t j00 = (size_t)y0 * W_LO + x0;
    const float invq = 1.0f / QMAX_LO;
#pragma unroll
    for (int c = 0; c < C_LO; ++c) {
      const float* p = lo + (size_t)c * HW_LO + j00;
      float a00 = qforward(p[0], QMAX_LO, invq);
      float a01 = qforward(p[1], QMAX_LO, invq);
      float a10 = qforward(p[W_LO], QMAX_LO, invq);
      float a11 = qforward(p[W_LO + 1], QMAX_LO, invq);
      r[C_HI + c] = w00 * a00 + w01 * a01 + w10 * a10 + w11 * a11;
    }
  }
  float4* o = (float4*)(out + (size_t)gid * 16);
  o[0] = make_float4(r[0], r[1], r[2], r[3]);
  o[1] = make_float4(r[4], r[5], r[6], r[7]);
  o[2] = make_float4(r[8], r[9], r[10], r[11]);
  o[3] = make_float4(r[12], r[13], r[14], r[15]);
}

extern "C" void kernel_launch(void* const* d_in, const int* in_sizes, int n_in,
                              void* d_out, int out_size, void* d_ws, size_t ws_size,
                              hipStream_t stream) {
  const float2* uv = (const float2*)d_in[0];
  const float*  hi = (const float*)d_in[1];
  const float*  lo = (const float*)d_in[2];
  float* out = (float*)d_out;
  const int B = in_sizes[0] / 2;

  const size_t hi_bytes = (size_t)HW_HI * 16;   // 64 MiB packed hi
  const size_t lo_bytes = (size_t)HW_LO * 4;    //  1 MiB packed lo
  const size_t need = hi_bytes + lo_bytes;

  if (ws_size >= need) {
    uint4*    hi8 = (uint4*)d_ws;
    unsigned* lo8 = (unsigned*)((char*)d_ws + hi_bytes);
    repack_hi_kernel<<<HW_HI / BLOCK, BLOCK, 0, stream>>>(hi, hi8);
    repack_lo_kernel<<<HW_LO / BLOCK, BLOCK, 0, stream>>>(lo, lo8);
    sample_packed_kernel<<<(B + BLOCK - 1) / BLOCK, BLOCK, 0, stream>>>(uv, hi8, lo8, out, B);
  } else {
    sample_naive_kernel<<<(B + BLOCK - 1) / BLOCK, BLOCK, 0, stream>>>(uv, hi, lo, out, B);
  }
}